// LSTM_55035710931638
// MI455X (gfx1250) — compile-verified
//
#include <hip/hip_runtime.h>

#define B 8
#define T 8192
#define D 20
#define H 8
#define G 32   // 4*H gate columns

typedef __attribute__((ext_vector_type(2))) float v2f;
typedef __attribute__((ext_vector_type(8))) float v8f;

// ---------------------------------------------------------------------------
// Kernel 1: pre0[b,t,:] = x[b,t,:] @ Wih0^T + bih0 + bhh0
// 65536 rows x 32 cols, K=20, fp32 WMMA 16x16x4. One wave per 16-row tile,
// 5 K-chunks x 2 N-tiles = 10 v_wmma_f32_16x16x4_f32 per wave.
// A 16x4 layout: VGPR0 = {lanes0-15: K=0, lanes16-31: K=2}, VGPR1 = K=1/K=3.
// B 4x16 layout mirrors A: per-lane N = lane&15, K-pair by lane>=16.
// ---------------------------------------------------------------------------
__global__ __launch_bounds__(256)
void lstm_pre_gemm(const float* __restrict__ x,
                   const float* __restrict__ Wih0,
                   const float* __restrict__ bih0,
                   const float* __restrict__ bhh0,
                   float* __restrict__ pre)
{
    const int lane = threadIdx.x & 31;
    const int wid  = threadIdx.x >> 5;
    const int row0 = (blockIdx.x * 8 + wid) * 16;   // 16-row tile of B*T rows
    const int m    = lane & 15;
    const int kk   = (lane >> 4) * 2;               // 0 or 2
    const float* xrow = x + (size_t)(row0 + m) * D;
    const int rbase = row0 + ((lane >> 4) * 8);     // C/D rows: m and m+8 halves

    #pragma unroll
    for (int nb = 0; nb < 2; ++nb) {
        const int col = nb * 16 + m;
        v8f acc = {};
        #pragma unroll
        for (int kc = 0; kc < D; kc += 4) {
            v2f a, bm;
            a.x  = xrow[kc + kk];
            a.y  = xrow[kc + kk + 1];
            bm.x = Wih0[col * D + kc + kk];          // B[k][n] = Wih0[n][k]
            bm.y = Wih0[col * D + kc + kk + 1];
            acc = __builtin_amdgcn_wmma_f32_16x16x4_f32(
                false, a, false, bm, (short)0, acc, false, false);
        }
        const float bias = bih0[col] + bhh0[col];
        #pragma unroll
        for (int r = 0; r < 8; ++r)
            pre[(size_t)(rbase + r) * G + col] = acc[r] + bias;
    }
}

// ---------------------------------------------------------------------------
// Wave-uniform lane broadcast: v_readlane_b32 -> SGPR, consumed as the scalar
// operand of v_fmac_f32. Keeps the serial recurrence off the DS (bpermute)
// pipeline. `lane` must be a compile-time/uniform value (it is: unrolled k).
// ---------------------------------------------------------------------------
__device__ __forceinline__ float bcast(float v, int lane) {
    return __uint_as_float(__builtin_amdgcn_readlane(__float_as_uint(v), lane));
}

// ---------------------------------------------------------------------------
// Activations: a = c1 + c2 * 1/(1+exp(s*g)) covers both sigmoid and tanh
// with per-lane constants (lanes 16-23 = cell gate -> tanh; others sigmoid).
// ---------------------------------------------------------------------------
__device__ __forceinline__ float act_eval(float g, float s, float c1, float c2) {
    float e = __expf(s * g);
    float u = 1.0f / (1.0f + e);
    return fmaf(c2, u, c1);
}
__device__ __forceinline__ float tanh_fast(float x) {
    float e = __expf(2.0f * x);                      // overflow-safe: inf -> 1
    return 1.0f - 2.0f / (e + 1.0f);
}

// ---------------------------------------------------------------------------
// Kernel 2: the serial 2-layer scan. One wave32 block per batch chain.
// Lane j owns gate column j (i:0-7 f:8-15 g:16-23 o:24-31); hidden state
// owned by lanes 0-7. h-broadcast = v_readlane (scalar FMA operand); gate
// regroup (lane-varying) = ds_bpermute via __shfl. Layer 1 runs at 1-step
// skew so both layers' chains overlap. pre0 stream register-prefetched 8
// steps ahead to hide L2 latency on the serial path.
// ---------------------------------------------------------------------------
__global__ __launch_bounds__(32)
void lstm_scan(const float* __restrict__ pre,
               const float* __restrict__ h0in, const float* __restrict__ c0in,
               const float* __restrict__ Whh0,
               const float* __restrict__ Wih1, const float* __restrict__ Whh1,
               const float* __restrict__ bih1, const float* __restrict__ bhh1,
               float* __restrict__ out, float* __restrict__ h1fin)
{
    const int b = blockIdx.x;
    const int j = threadIdx.x;
    const int m = j & 7;
    const int grp = j >> 3;

    float w0[H], wi1[H], wh1[H];
    #pragma unroll
    for (int k = 0; k < H; ++k) {
        w0[k]  = Whh0[j * H + k];
        wi1[k] = Wih1[j * H + k];
        wh1[k] = Whh1[j * H + k];
    }
    const float bias1 = bih1[j] + bhh1[j];
    const float s  = (grp == 2) ?  2.0f : -1.0f;
    const float c1 = (grp == 2) ?  1.0f :  0.0f;
    const float c2 = (grp == 2) ? -2.0f :  1.0f;

    float h0s = h0in[b * H + m];
    float c0s = c0in[b * H + m];
    float h1s = h0in[B * H + b * H + m];
    float c1s = c0in[B * H + b * H + m];

    const float* prb = pre + (size_t)b * T * G + j;

    // t = 0: layer 0 only
    {
        float g = prb[0];
        #pragma unroll
        for (int k = 0; k < H; ++k) g = fmaf(bcast(h0s, k), w0[k], g);
        float a = act_eval(g, s, c1, c2);
        float i_ = __shfl(a, m),      f_ = __shfl(a, m + 8),
              g_ = __shfl(a, m + 16), o_ = __shfl(a, m + 24);
        c0s = fmaf(f_, c0s, i_ * g_);
        h0s = o_ * tanh_fast(c0s);
    }

    constexpr int PF = 8;
    float pbuf[PF];
    #pragma unroll
    for (int u = 0; u < PF; ++u) pbuf[u] = prb[(size_t)(1 + u) * G];

    // iterations t = 1..T : layer0 at time t (t<T), layer1 at time t-1
    for (int tb = 1; tb <= T; tb += PF) {
        #pragma unroll
        for (int u = 0; u < PF; ++u) {
            const int t = tb + u;
            const float hA = h0s;                       // h0 output at t-1
            // ---- layer 1, time t-1 (independent of layer-0 step below) ----
            float gb = bias1;
            #pragma unroll
            for (int k = 0; k < H; ++k) gb = fmaf(bcast(hA,  k), wi1[k], gb);
            #pragma unroll
            for (int k = 0; k < H; ++k) gb = fmaf(bcast(h1s, k), wh1[k], gb);
            float a1 = act_eval(gb, s, c1, c2);
            // ---- layer 0, time t ----
            if (t < T) {                                 // uniform branch
                float ga = pbuf[u];
                #pragma unroll
                for (int k = 0; k < H; ++k) ga = fmaf(bcast(hA, k), w0[k], ga);
                float a0 = act_eval(ga, s, c1, c2);
                float i0 = __shfl(a0, m),      f0 = __shfl(a0, m + 8),
                      g0 = __shfl(a0, m + 16), o0 = __shfl(a0, m + 24);
                c0s = fmaf(f0, c0s, i0 * g0);
                h0s = o0 * tanh_fast(c0s);
                int tp = t + PF; if (tp > T - 1) tp = T - 1;
                pbuf[u] = prb[(size_t)tp * G];           // refill PF slot
            }
            float i1 = __shfl(a1, m),      f1 = __shfl(a1, m + 8),
                  g1 = __shfl(a1, m + 16), o1g = __shfl(a1, m + 24);
            c1s = fmaf(f1, c1s, i1 * g1);
            h1s = o1g * tanh_fast(c1s);
        }
    }

    // output layout: o2[2] | hT[2,8,8] | cT[2,8,8]
    if (j < H) {
        out[2 + b * H + m]             = h0s;   // hT layer 0
        out[2 + B * H + b * H + m]     = h1s;   // hT layer 1
        out[2 + 2 * B * H + b * H + m] = c0s;   // cT layer 0
        out[2 + 3 * B * H + b * H + m] = c1s;   // cT layer 1
        h1fin[b * H + m] = h1s;
    }
}

// ---------------------------------------------------------------------------
// Kernel 3: head. Only o1[:,3] is consumed by the reference's quirk, so just
// compute that column. ~100 flops -> single thread.
// ---------------------------------------------------------------------------
__global__ void lstm_head(const float* __restrict__ h1fin,
                          const float* __restrict__ Wfc, const float* __restrict__ bfc,
                          const float* __restrict__ Wi2h, const float* __restrict__ bi2h,
                          float* __restrict__ out)
{
    if (threadIdx.x == 0 && blockIdx.x == 0) {
        float r[B];
        #pragma unroll
        for (int bb = 0; bb < B; ++bb) {
            float o1 = bfc[3];
            #pragma unroll
            for (int mm = 0; mm < H; ++mm)
                o1 = fmaf(fmaxf(h1fin[bb * H + mm], 0.0f), Wfc[3 * H + mm], o1);
            r[bb] = fmaxf(o1, 0.0f);
        }
        #pragma unroll
        for (int jj = 0; jj < 2; ++jj) {
            float o2 = bi2h[jj];
            #pragma unroll
            for (int bb = 0; bb < B; ++bb)
                o2 = fmaf(r[bb], Wi2h[jj * B + bb], o2);
            out[jj] = o2;
        }
    }
}

// ---------------------------------------------------------------------------
extern "C" void kernel_launch(void* const* d_in, const int* in_sizes, int n_in,
                              void* d_out, int out_size, void* d_ws, size_t ws_size,
                              hipStream_t stream)
{
    const float* x    = (const float*)d_in[0];
    const float* h0   = (const float*)d_in[1];
    const float* c0   = (const float*)d_in[2];
    const float* Wih0 = (const float*)d_in[3];
    const float* Whh0 = (const float*)d_in[4];
    const float* bih0 = (const float*)d_in[5];
    const float* bhh0 = (const float*)d_in[6];
    const float* Wih1 = (const float*)d_in[7];
    const float* Whh1 = (const float*)d_in[8];
    const float* bih1 = (const float*)d_in[9];
    const float* bhh1 = (const float*)d_in[10];
    const float* Wfc  = (const float*)d_in[11];
    const float* bfc  = (const float*)d_in[12];
    const float* Wi2h = (const float*)d_in[13];
    const float* bi2h = (const float*)d_in[14];
    float* out = (float*)d_out;

    float* h1fin = (float*)d_ws;                     // 64 floats
    float* pre   = (float*)((char*)d_ws + 256);      // B*T*G floats = 8 MB

    lstm_pre_gemm<<<(B * T) / (16 * 8), 256, 0, stream>>>(x, Wih0, bih0, bhh0, pre);
    lstm_scan<<<B, 32, 0, stream>>>(pre, h0, c0, Whh0, Wih1, Whh1, bih1, bhh1,
                                    out, h1fin);
    lstm_head<<<1, 32, 0, stream>>>(h1fin, Wfc, bfc, Wi2h, bi2h, out);
}